// RPNProposal_2705829396907
// MI455X (gfx1250) — compile-verified
//
#include <hip/hip_runtime.h>
#include <hip/hip_bf16.h>
#include <stdint.h>

#define B      8
#define H      64
#define W      64
#define KA     9
#define NPB    (H*W*KA)       /* 36864 boxes per batch */
#define TOPN   2000
#define NSORT  2048           /* padded to power of two for bitonic sort */
#define NW     (NSORT/32)     /* 64 mask words per row */
#define NBINS  4096
#define STRIDE 16.0f
#define IMG    1024.0f
#define MIN_SZ 0.01f
#define IOU_TH 0.7f
#define EPS_F  1e-7f

typedef float v2f __attribute__((ext_vector_type(2)));
typedef float v8f __attribute__((ext_vector_type(8)));

// ---------------------------------------------------------------- decode ----
__global__ void k_decode(const float* __restrict__ scores,
                         const float* __restrict__ offsets,
                         const float* __restrict__ anchors,
                         float* __restrict__ boxesDec,   // [B*NPB*4]
                         float* __restrict__ scoresDec,  // [B*NPB]
                         unsigned* __restrict__ hist)    // [B*NBINS]
{
    int g = blockIdx.x * blockDim.x + threadIdx.x;
    if (g >= B * NPB) return;
    int b = g / NPB;
    int r = g % NPB;
    int k = r % KA;
    int x = (r / KA) % W;
    int y = r / (KA * W);

    float sc = scores[g];  // scores flat layout == g
    const float* off = offsets + (((size_t)(b * H + y) * W + x) * (4 * KA) + 4 * k);
    float aw = anchors[2 * k], ah = anchors[2 * k + 1];

    float cx = (aw * off[0] + (x + 0.5f) * STRIDE) / IMG;
    float cy = (ah * off[1] + (y + 0.5f) * STRIDE) / IMG;
    float pw = __expf(off[2]) * aw / IMG;
    float ph = __expf(off[3]) * ah / IMG;

    float x1 = fminf(fmaxf(cx - 0.5f * pw, 0.f), 1.f);
    float y1 = fminf(fmaxf(cy - 0.5f * ph, 0.f), 1.f);
    float x2 = fminf(fmaxf(cx + 0.5f * pw, 0.f), 1.f);
    float y2 = fminf(fmaxf(cy + 0.5f * ph, 0.f), 1.f);
    float s  = fminf(fmaxf(sc, 0.f), 1.f);
    float vm = ((x2 - x1) > MIN_SZ && (y2 - y1) > MIN_SZ) ? 1.f : 0.f;
    x1 *= vm; y1 *= vm; x2 *= vm; y2 *= vm; s *= vm;

    ((float4*)boxesDec)[g] = make_float4(x1, y1, x2, y2);
    scoresDec[g] = s;

    int bin = (int)(s * (float)NBINS);
    if (bin > NBINS - 1) bin = NBINS - 1;
    atomicAdd(&hist[b * NBINS + bin], 1u);
}

// ------------------------------------------------------ top-k threshold -----
__global__ void k_thresh(const unsigned* __restrict__ hist,
                         int* __restrict__ tInfo)   // [B*4]: T, G, R
{
    __shared__ unsigned h[NBINS];
    int b = blockIdx.x;
    for (int t = threadIdx.x; t < NBINS; t += blockDim.x) h[t] = hist[b * NBINS + t];
    __syncthreads();
    if (threadIdx.x == 0) {
        int c = 0, T = 0, G = 0;
        for (int t = NBINS - 1; t >= 0; --t) {
            int ht = (int)h[t];
            if (c + ht >= TOPN) { T = t; G = c; break; }
            c += ht;
        }
        tInfo[b * 4 + 0] = T;
        tInfo[b * 4 + 1] = G;
        tInfo[b * 4 + 2] = TOPN - G;
    }
}

// --------------------------------------------------------- compaction -------
__global__ void k_compact(const float* __restrict__ scoresDec,
                          const int* __restrict__ tInfo,
                          unsigned* __restrict__ cntHi,
                          unsigned* __restrict__ cntEq,
                          unsigned long long* __restrict__ candKeys)
{
    int g = blockIdx.x * blockDim.x + threadIdx.x;
    if (g >= B * NPB) return;
    int b = g / NPB;
    int idx = g % NPB;
    float s = scoresDec[g];
    int bin = (int)(s * (float)NBINS);
    if (bin > NBINS - 1) bin = NBINS - 1;
    int T = tInfo[b * 4 + 0], G = tInfo[b * 4 + 1], R = tInfo[b * 4 + 2];
    // key: descending score, tie-break ascending index
    unsigned long long key = ((unsigned long long)__float_as_uint(s) << 32)
                           | (unsigned)(0xFFFFFFFFu - (unsigned)idx);
    if (bin > T) {
        unsigned slot = atomicAdd(&cntHi[b], 1u);     // slot < G <= TOPN
        candKeys[(size_t)b * NSORT + slot] = key;
    } else if (bin == T) {
        unsigned e = atomicAdd(&cntEq[b], 1u);
        if (e < (unsigned)R) candKeys[(size_t)b * NSORT + G + e] = key;
    }
}

// ------------------------------------------ bitonic sort + gather top boxes -
__global__ void __launch_bounds__(1024)
k_sort_gather(const unsigned long long* __restrict__ candKeys,
              const float* __restrict__ boxesDec,
              float* __restrict__ topBoxes,   // [B*NSORT*4]
              float* __restrict__ areas)      // [B*NSORT]
{
    __shared__ unsigned long long key[NSORT];
    int b = blockIdx.x;
    int t = threadIdx.x;
    key[t]        = candKeys[(size_t)b * NSORT + t];
    key[t + 1024] = candKeys[(size_t)b * NSORT + t + 1024];
    __syncthreads();
    for (int k = 2; k <= NSORT; k <<= 1) {
        for (int j = k >> 1; j > 0; j >>= 1) {
            int i = 2 * j * (t / j) + (t % j);
            int p = i + j;
            unsigned long long a = key[i], c = key[p];
            bool desc = ((i & k) == 0);
            if ((a < c) == desc) { key[i] = c; key[p] = a; }
            __syncthreads();
        }
    }
    for (int rr = 0; rr < 2; ++rr) {
        int r = t + rr * 1024;
        float4 bx = make_float4(0.f, 0.f, 0.f, 0.f);
        if (r < TOPN) {
            unsigned idx = 0xFFFFFFFFu - (unsigned)(key[r] & 0xFFFFFFFFull);
            bx = ((const float4*)boxesDec)[(size_t)b * NPB + idx];
        }
        ((float4*)topBoxes)[(size_t)b * NSORT + r] = bx;
        areas[(size_t)b * NSORT + r] = (bx.z - bx.x) * (bx.w - bx.y);
    }
}

// --------------------------- pairwise IoU suppression bitmask (WMMA tiles) --
// One wave owns a 16x32 (i,j) tile == one 32-bit mask word per row, so plain
// stores (no atomics). Each 16-col half uses V_WMMA_F32_16X16X4_F32 as a
// rank-2 product to form the tile of (area_i + area_j):
//   A[i][:] = [area_i, 1, 0, 0],  B[:][j] = [1, area_j, 0, 0]^T
// The IoU threshold test is division-free: denom > 0 always, so
//   inter/denom > TH  <=>  inter > TH * denom.
__global__ void k_iou_mask(const float* __restrict__ topBoxes,
                           const float* __restrict__ areas,
                           unsigned* __restrict__ maskMat)  // [B*NSORT*NW]
{
    const int TI = NSORT / 16;   // 128 row tiles
    const int TJ = NSORT / 32;   // 64 column words
    int wavesPerBlk = blockDim.x >> 5;
    int waveId = blockIdx.x * wavesPerBlk + (threadIdx.x >> 5);
    int lane = threadIdx.x & 31;
    int total = B * TI * TJ;
    if (waveId >= total) return;             // wave-uniform
    int b   = waveId / (TI * TJ);
    int rem = waveId % (TI * TJ);
    int ti  = rem / TJ;
    int tw  = rem % TJ;
    int i0 = ti * 16;
    int j0 = tw * 32;
    if (j0 + 31 <= i0) return;               // lower triangle: no j > i pairs

    bool hilane = lane >= 16;
    int  l = lane & 15;
    float areaI = hilane ? 0.f : areas[(size_t)b * NSORT + i0 + l];
    const float4* tb = (const float4*)topBoxes;

    unsigned wordLo[8], wordHi[8];           // rows i0+r and i0+r+8, word tw

    #pragma unroll
    for (int hh = 0; hh < 2; ++hh) {
        int jb = j0 + hh * 16;
        // A: 16x4 f32 layout (lanes 0-15 -> K=0..1, lanes 16-31 -> K=2..3)
        v2f av; av[0] = areaI;               av[1] = hilane ? 0.f : 1.0f;
        // B: 4x16 f32 layout (lanes 0-15 -> K=0..1 rows, lanes 16-31 -> K=2..3)
        v2f bv; bv[0] = hilane ? 0.f : 1.0f;
                bv[1] = hilane ? 0.f : areas[(size_t)b * NSORT + jb + l];
        v8f cv = {};
        cv = __builtin_amdgcn_wmma_f32_16x16x4_f32(
                /*neg_a=*/false, av, /*neg_b=*/false, bv,
                /*c_mod=*/(short)0, cv, /*reuse_a=*/false, /*reuse_b=*/false);

        float4 bj = tb[(size_t)b * NSORT + jb + l];
        int jcol = jb + l;
        #pragma unroll
        for (int r = 0; r < 8; ++r) {
            int i = i0 + r + (hilane ? 8 : 0);   // D layout: lane<16 -> M=r, else M=r+8
            float4 bi = tb[(size_t)b * NSORT + i];
            float xx1 = fmaxf(bi.x, bj.x), yy1 = fmaxf(bi.y, bj.y);
            float xx2 = fminf(bi.z, bj.z), yy2 = fminf(bi.w, bj.w);
            float inter = fmaxf(xx2 - xx1, 0.f) * fmaxf(yy2 - yy1, 0.f);
            // denom = area_i + area_j - inter + eps > 0 always
            bool bit = (inter > IOU_TH * (cv[r] - inter + EPS_F)) && (jcol > i);
            // direct wave32 ballot: no bool->int->cmp roundtrip
            unsigned bal = __builtin_amdgcn_ballot_w32(bit);
            if (hh == 0) { wordLo[r] = bal & 0xFFFFu;           wordHi[r] = bal >> 16; }
            else         { wordLo[r] |= (bal & 0xFFFFu) << 16;  wordHi[r] |= (bal >> 16) << 16; }
        }
    }
    // single owner per (row, word): plain stores, no atomics
    #pragma unroll
    for (int r = 0; r < 8; ++r) {
        if (lane == 0)
            maskMat[((size_t)b * NSORT + i0 + r) * NW + tw] = wordLo[r];
        else if (lane == 16)
            maskMat[((size_t)b * NSORT + i0 + r + 8) * NW + tw] = wordHi[r];
    }
}

// --------------------------------------------- serial NMS scan + emit -------
__global__ void __launch_bounds__(64)
k_nms(const float* __restrict__ topBoxes,
      const unsigned* __restrict__ maskMat,
      float* __restrict__ out)
{
    __shared__ unsigned remv[NW];
    __shared__ int order[TOPN];
    __shared__ int cnt;
    int b = blockIdx.x;
    int t = threadIdx.x;
    remv[t] = 0u;
    if (t == 0) cnt = 0;
    __syncthreads();
    for (int i = 0; i < TOPN; ++i) {
        // speculative prefetch of a row we'll very likely need soon
        int pf = i + 8;
        if (pf < TOPN)
            __builtin_prefetch(&maskMat[((size_t)b * NSORT + pf) * NW + t], 0, 1);
        bool supp = (remv[i >> 5] >> (i & 31)) & 1u;   // uniform across block
        __syncthreads();
        if (!supp) {
            remv[t] |= maskMat[((size_t)b * NSORT + i) * NW + t];
            if (t == 0) order[cnt++] = i;
        }
        __syncthreads();
    }
    int c = cnt;
    const float4* tb = (const float4*)topBoxes;
    float4* o = (float4*)out;
    for (int r = t; r < TOPN; r += 64) {
        float4 v = make_float4(0.f, 0.f, 0.f, 0.f);
        if (r < c) v = tb[(size_t)b * NSORT + order[r]];
        o[(size_t)b * TOPN + r] = v;                  // all rows written
    }
}

// ----------------------------------------------------------------------------
extern "C" void kernel_launch(void* const* d_in, const int* in_sizes, int n_in,
                              void* d_out, int out_size, void* d_ws, size_t ws_size,
                              hipStream_t stream) {
    const float* scores  = (const float*)d_in[0];
    const float* offsets = (const float*)d_in[1];
    const float* anchors = (const float*)d_in[2];

    char* ws = (char*)d_ws;
    size_t off = 0;
    auto take = [&](size_t bytes) -> void* {
        void* p = ws + off;
        off += (bytes + 255) & ~(size_t)255;
        return p;
    };
    float*              boxesDec  = (float*)take((size_t)B * NPB * 4 * sizeof(float));
    float*              scoresDec = (float*)take((size_t)B * NPB * sizeof(float));
    unsigned*           hist      = (unsigned*)take((size_t)B * NBINS * sizeof(unsigned));
    int*                tInfo     = (int*)take((size_t)B * 4 * sizeof(int));
    unsigned*           cntHi     = (unsigned*)take((size_t)B * sizeof(unsigned));
    unsigned*           cntEq     = (unsigned*)take((size_t)B * sizeof(unsigned));
    unsigned long long* candKeys  = (unsigned long long*)take((size_t)B * NSORT * sizeof(unsigned long long));
    float*              topBoxes  = (float*)take((size_t)B * NSORT * 4 * sizeof(float));
    float*              areas     = (float*)take((size_t)B * NSORT * sizeof(float));
    unsigned*           maskMat   = (unsigned*)take((size_t)B * NSORT * NW * sizeof(unsigned));

    // zero the regions that accumulate (graph-capture-safe memset nodes)
    hipMemsetAsync(hist,     0, (size_t)B * NBINS * sizeof(unsigned), stream);
    hipMemsetAsync(cntHi,    0, (size_t)B * sizeof(unsigned), stream);
    hipMemsetAsync(cntEq,    0, (size_t)B * sizeof(unsigned), stream);
    hipMemsetAsync(candKeys, 0, (size_t)B * NSORT * sizeof(unsigned long long), stream);
    hipMemsetAsync(maskMat,  0, (size_t)B * NSORT * NW * sizeof(unsigned), stream);

    int n = B * NPB;
    k_decode<<<(n + 255) / 256, 256, 0, stream>>>(scores, offsets, anchors,
                                                  boxesDec, scoresDec, hist);
    k_thresh<<<B, 256, 0, stream>>>(hist, tInfo);
    k_compact<<<(n + 255) / 256, 256, 0, stream>>>(scoresDec, tInfo, cntHi, cntEq, candKeys);
    k_sort_gather<<<B, 1024, 0, stream>>>(candKeys, boxesDec, topBoxes, areas);

    int totalWaves = B * (NSORT / 16) * (NSORT / 32);   // 65536 16x32 tiles
    int blocks = (totalWaves + 7) / 8;                  // 8 waves per 256-thread block
    k_iou_mask<<<blocks, 256, 0, stream>>>(topBoxes, areas, maskMat);

    k_nms<<<B, 64, 0, stream>>>(topBoxes, maskMat, (float*)d_out);
}